// GraphSAGE_42941083025882
// MI455X (gfx1250) — compile-verified
//
#include <hip/hip_runtime.h>
#include <cstddef>

typedef __attribute__((ext_vector_type(2))) float v2f;
typedef __attribute__((ext_vector_type(8))) float v8f;

#define Bn 2
#define Cn 192
#define Nn 16384
#define Kn 16
#define On 384
#define PITCH 20  // floats per LDS column: 16 data + 4 pad -> conflict-free, 8B aligned

// ---- gfx1250 async global->LDS path (gated; falls back to sync copies) ----
#if defined(__has_builtin)
#if __has_builtin(__builtin_amdgcn_global_load_async_to_lds_b64) && \
    __has_builtin(__builtin_amdgcn_global_load_async_to_lds_b32)
#define USE_ASYNC 1
#endif
#endif
#ifndef USE_ASYNC
#define USE_ASYNC 0
#endif

#if USE_ASYNC
typedef int v2i_t __attribute__((vector_size(8)));
#define AS_G __attribute__((address_space(1)))
#define AS_L __attribute__((address_space(3)))
#define ASYNC_B64(g, l)                                                \
  __builtin_amdgcn_global_load_async_to_lds_b64((AS_G v2i_t*)(g),      \
                                                (AS_L v2i_t*)(l), 0, 0)
#define ASYNC_B32(g, l)                                          \
  __builtin_amdgcn_global_load_async_to_lds_b32((AS_G int*)(g),  \
                                                (AS_L int*)(l), 0, 0)
#define ASYNC_WAIT(n) asm volatile("s_wait_asynccnt " #n ::: "memory")
#else
#define ASYNC_B64(g, l) (*(v2f*)(l) = *(const v2f*)(g))
#define ASYNC_B32(g, l) (*(float*)(l) = *(const float*)(g))
#define ASYNC_WAIT(n)
#endif

// ---------------------------------------------------------------------------
// Kernel A: transpose x [B,C,N] -> xt [B,N,C] so each gathered column is a
// contiguous 768B run for GEMM1's neighbor gather.
// ---------------------------------------------------------------------------
__global__ __launch_bounds__(256) void xpose_kernel(const float* __restrict__ x,
                                                    float* __restrict__ xt) {
  __shared__ float tile[32][33];
  const int b  = blockIdx.z;
  const int c0 = blockIdx.x * 32;
  const int n0 = blockIdx.y * 32;
  const int tx = threadIdx.x, ty = threadIdx.y;
#pragma unroll
  for (int i = 0; i < 32; i += 8)
    tile[ty + i][tx] =
        x[(size_t)b * Cn * Nn + (size_t)(c0 + ty + i) * Nn + (n0 + tx)];
  __syncthreads();
#pragma unroll
  for (int i = 0; i < 32; i += 8)
    xt[(size_t)b * Nn * Cn + (size_t)(n0 + ty + i) * Cn + (c0 + tx)] =
        tile[tx][ty + i];
}

// ---------------------------------------------------------------------------
// Kernel B: gather + GEMM1 (w1 192x192) + bias + ReLU + max over K neighbors.
// WG = 8 nodes * 16 neighbors = 128 columns, 8 waves (256 threads).
// Wave w owns col-tile w (= node w) x all 12 M-tiles -> each B-fragment
// ds_load feeds 12 WMMAs. 16-channel K-slabs double-buffered in LDS with
// async global->LDS copies (4 b64 per thread per slab, uniform per wave).
// h (402 MB logical) never leaves accumulator VGPRs.
// ---------------------------------------------------------------------------
__global__ __launch_bounds__(256) void gemm1_kernel(
    const float* __restrict__ xt, const long long* __restrict__ edge,
    const float* __restrict__ w1, const float* __restrict__ b1,
    float* __restrict__ hmax) {
  __shared__ float buf[2][128 * PITCH];

  const int b     = blockIdx.y;
  const int node0 = blockIdx.x * 8;
  const int tid   = (int)threadIdx.x;
  const int wave  = tid >> 5;
  const int lane  = tid & 31;
  const int lrow  = lane & 15;
  const int lhi   = lane >> 4;

  const float* xtb = xt + (size_t)b * Nn * Cn;

  // Fixed per-thread gather assignment: 4 b64 units of the 128x16 slab.
  const float* gsrc[4];
  int loff[4];
#pragma unroll
  for (int j = 0; j < 4; ++j) {
    const int u    = tid + 256 * j;  // 0..1023
    const int col  = u >> 3;         // 0..127  (node_local*16 + k)
    const int pair = u & 7;          // which 2-float unit of the 16-ch slab
    const long long idx =
        edge[((size_t)b * Nn + (node0 + (col >> 4))) * Kn + (col & 15)];
    gsrc[j] = xtb + (size_t)((int)idx) * Cn + pair * 2;
    loff[j] = col * PITCH + pair * 2;
  }

  v8f acc[12];
#pragma unroll
  for (int mt = 0; mt < 12; ++mt) acc[mt] = (v8f)0.0f;

  // Preload slab 0 into buffer 0.
#pragma unroll
  for (int j = 0; j < 4; ++j) ASYNC_B64(gsrc[j], &buf[0][loff[j]]);

  for (int s = 0; s < Cn / 16; ++s) {
    const int cb = s * 16;
    if (s + 1 < Cn / 16) {  // issue slab s+1, then wait for slab s (in-order)
      float* dst = buf[(s + 1) & 1];
#pragma unroll
      for (int j = 0; j < 4; ++j) ASYNC_B64(gsrc[j] + cb + 16, &dst[loff[j]]);
      ASYNC_WAIT(0x4);
    } else {
      ASYNC_WAIT(0x0);
    }
    __syncthreads();
    const float* bsl = buf[s & 1];
#pragma unroll
    for (int kk = 0; kk < 4; ++kk) {
      const v2f bb =
          *(const v2f*)&bsl[(wave * 16 + lrow) * PITCH + kk * 4 + 2 * lhi];
#pragma unroll
      for (int mt = 0; mt < 12; ++mt) {
        const v2f a = *(const v2f*)(w1 + (size_t)(mt * 16 + lrow) * Cn + cb +
                                    kk * 4 + 2 * lhi);
        acc[mt] = __builtin_amdgcn_wmma_f32_16x16x4_f32(
            false, a, false, bb, (short)0, acc[mt], false, false);
      }
    }
    __syncthreads();
  }

  // Per tile: max over the 16 columns (= this node's neighbors), then
  // bias + ReLU (commute with max since bias is per-row), store hmax[B,C,N].
  const int node = node0 + wave;
#pragma unroll
  for (int mt = 0; mt < 12; ++mt) {
#pragma unroll
    for (int v = 0; v < 8; ++v) {
      float m = acc[mt][v];
#pragma unroll
      for (int off = 8; off; off >>= 1) m = fmaxf(m, __shfl_xor(m, off, 16));
      if (lrow == 0) {
        const int row = mt * 16 + v + 8 * lhi;
        hmax[(size_t)b * Cn * Nn + (size_t)row * Nn + node] =
            fmaxf(m + b1[row], 0.0f);
      }
    }
  }
}

// ---------------------------------------------------------------------------
// Kernel C: GEMM2 (w2 384x384) over cat = [x ; hmax] + bias + ReLU.
// WG = 128 nodes, 16 waves (512 threads); wave w -> col-tile (w&7),
// M-tile group (w>>3)*12..+11. Slabs double-buffered via async b32 copies
// (4 per thread, uniform). Both cat halves are channel-major -> coalesced.
// ---------------------------------------------------------------------------
__global__ __launch_bounds__(512) void gemm2_kernel(
    const float* __restrict__ x, const float* __restrict__ hmax,
    const float* __restrict__ w2, const float* __restrict__ b2,
    float* __restrict__ out) {
  __shared__ float buf[2][128 * PITCH];

  const int b    = blockIdx.y;
  const int n0   = blockIdx.x * 128;
  const int tid  = (int)threadIdx.x;
  const int wave = tid >> 5;
  const int lane = tid & 31;
  const int lrow = lane & 15;
  const int lhi  = lane >> 4;
  const int ct   = wave & 7;   // col tile
  const int mg   = wave >> 3;  // 0/1 -> rows mg*192 .. +191

  const float* xb = x + (size_t)b * Cn * Nn;
  const float* hb = hmax + (size_t)b * Cn * Nn;

  int col_[4], ch_[4];
#pragma unroll
  for (int j = 0; j < 4; ++j) {
    const int i = tid + 512 * j;  // 0..2047
    col_[j] = i & 127;
    ch_[j]  = i >> 7;  // 0..15
  }

  v8f acc[12];
#pragma unroll
  for (int mt = 0; mt < 12; ++mt) acc[mt] = (v8f)0.0f;

  // Preload slab 0.
  {
    const float* src = xb;  // slab 0 is x channels 0..15
#pragma unroll
    for (int j = 0; j < 4; ++j)
      ASYNC_B32(src + (size_t)ch_[j] * Nn + n0 + col_[j],
                &buf[0][col_[j] * PITCH + ch_[j]]);
  }

  for (int s = 0; s < (2 * Cn) / 16; ++s) {
    const int cb = s * 16;
    if (s + 1 < (2 * Cn) / 16) {
      const int cb1 = (s + 1) * 16;
      const float* src = (cb1 < Cn) ? (xb + (size_t)cb1 * Nn)
                                    : (hb + (size_t)(cb1 - Cn) * Nn);
      float* dst = buf[(s + 1) & 1];
#pragma unroll
      for (int j = 0; j < 4; ++j)
        ASYNC_B32(src + (size_t)ch_[j] * Nn + n0 + col_[j],
                  &dst[col_[j] * PITCH + ch_[j]]);
      ASYNC_WAIT(0x4);
    } else {
      ASYNC_WAIT(0x0);
    }
    __syncthreads();
    const float* bsl = buf[s & 1];
#pragma unroll
    for (int kk = 0; kk < 4; ++kk) {
      const v2f bb =
          *(const v2f*)&bsl[(ct * 16 + lrow) * PITCH + kk * 4 + 2 * lhi];
#pragma unroll
      for (int mt = 0; mt < 12; ++mt) {
        const int m = (mg * 12 + mt) * 16 + lrow;
        const v2f a = *(const v2f*)(w2 + (size_t)m * (2 * Cn) + cb + kk * 4 +
                                    2 * lhi);
        acc[mt] = __builtin_amdgcn_wmma_f32_16x16x4_f32(
            false, a, false, bb, (short)0, acc[mt], false, false);
      }
    }
    __syncthreads();
  }

  // Epilogue: bias + ReLU, coalesced stores to out[B,OUT,N].
#pragma unroll
  for (int mt = 0; mt < 12; ++mt) {
#pragma unroll
    for (int v = 0; v < 8; ++v) {
      const int o = (mg * 12 + mt) * 16 + v + 8 * lhi;
      const int n = n0 + ct * 16 + lrow;
      out[(size_t)b * On * Nn + (size_t)o * Nn + n] =
          fmaxf(acc[mt][v] + b2[o], 0.0f);
    }
  }
}

// ---------------------------------------------------------------------------
extern "C" void kernel_launch(void* const* d_in, const int* in_sizes, int n_in,
                              void* d_out, int out_size, void* d_ws,
                              size_t ws_size, hipStream_t stream) {
  const float*     x    = (const float*)d_in[0];
  const long long* edge = (const long long*)d_in[1];  // int64 indices
  const float*     w1   = (const float*)d_in[2];
  const float*     b1   = (const float*)d_in[3];
  const float*     w2   = (const float*)d_in[4];
  const float*     b2   = (const float*)d_in[5];
  float*           out  = (float*)d_out;

  float* xt   = (float*)d_ws;               // [B,N,C]  25.2 MB
  float* hmax = xt + (size_t)Bn * Nn * Cn;  // [B,C,N]  25.2 MB

  dim3 gA(Cn / 32, Nn / 32, Bn);
  xpose_kernel<<<gA, dim3(32, 8), 0, stream>>>(x, xt);

  dim3 gB(Nn / 8, Bn);
  gemm1_kernel<<<gB, 256, 0, stream>>>(xt, edge, w1, b1, hmax);

  dim3 gC(Nn / 128, Bn);
  gemm2_kernel<<<gC, 512, 0, stream>>>(x, hmax, w2, b2, out);
}